// Voxelizer_22247930593310
// MI455X (gfx1250) — compile-verified
//
#include <hip/hip_runtime.h>
#include <hip/hip_bf16.h>

// ---------------------------------------------------------------------------
// Voxelizer: [B, N, 3] float32 points -> [B, 20, 512, 512] occupancy (float).
// Pure memory-bound scatter; no matrix math. Strategy:
//   1) vox_zero:    B128 stores to zero the 335.5 MB grid.
//   2) vox_scatter: each thread loads 4 points as 3x float4 (48B, coalesced,
//                   TH=NT so the one-shot stream doesn't pollute L2), computes
//                   floor-quantized clipped indices exactly like the JAX ref,
//                   and stores 1.0f for valid points (benign same-value race).
// ---------------------------------------------------------------------------

typedef float v4f __attribute__((ext_vector_type(4)));

#define VX_XMIN  (-51.2f)
#define VX_XMAX  ( 51.2f)
#define VX_YMIN  (-51.2f)
#define VX_YMAX  ( 51.2f)
#define VX_ZMIN  ( -3.0f)
#define VX_ZMAX  (  1.0f)
#define VX_STEP  (  0.2f)
#define VX_D     20
#define VX_H     512
#define VX_W     512

__global__ void vox_zero(float* __restrict__ out, int n4, int total) {
    int i = blockIdx.x * blockDim.x + threadIdx.x;
    v4f* __restrict__ out4 = (v4f*)out;
    if (i < n4) {
        v4f z = (v4f)0.0f;
        out4[i] = z;                      // global_store_b128
    }
    // Scalar tail (out_size % 4), handled by thread 0 only (normally empty).
    if (i == 0) {
        for (int r = n4 * 4; r < total; ++r) out[r] = 0.0f;
    }
}

__device__ __forceinline__ void vox_one_point(float x, float y, float z,
                                              int bidx, float* __restrict__ out) {
    // Strict in-range filter, identical to the reference's bitwise-and chain.
    bool valid = (x > VX_XMIN) && (x < VX_XMAX) &&
                 (y > VX_YMIN) && (y < VX_YMAX) &&
                 (z > VX_ZMIN) && (z < VX_ZMAX);
    // floor((v - min)/step) (y: floor((max - v)/step)) in fp32, exactly as jnp.floor(.../STEP): one
    // round-down convert. Then clip so boundary fp-rounding (e.g. index 512)
    // lands on the last voxel, matching jnp.clip.
    int iz = __float2int_rd((z - VX_ZMIN) / VX_STEP);
    int iy = __float2int_rd((VX_YMAX - y) / VX_STEP);
    int ix = __float2int_rd((x - VX_XMIN) / VX_STEP);
    iz = min(max(iz, 0), VX_D - 1);
    iy = min(max(iy, 0), VX_H - 1);
    ix = min(max(ix, 0), VX_W - 1);
    if (valid) {
        size_t o = ((((size_t)bidx * VX_D + iz) * VX_H) + iy) * (size_t)VX_W + ix;
        out[o] = 1.0f;                    // scatter-max(True) == store 1.0f
    }
}

__global__ void vox_scatter(const float* __restrict__ pts,
                            float* __restrict__ out,
                            int npts, int N) {
    int t    = blockIdx.x * blockDim.x + threadIdx.x;
    int base = t * 4;                     // first flat point index
    if (base >= npts) return;

    // Batch id: one divide per thread; the 4-point window crosses at most
    // one batch boundary (N >= 4).
    int b0  = base / N;
    int rem = base - b0 * N;

    if (base + 4 <= npts) {
        // 48 bytes per lane as 3x b128 loads, non-temporal (read-once stream).
        const v4f* __restrict__ p4 = (const v4f*)(pts) + (size_t)t * 3;
        v4f a = __builtin_nontemporal_load(p4 + 0);
        v4f b = __builtin_nontemporal_load(p4 + 1);
        v4f c = __builtin_nontemporal_load(p4 + 2);
        float xs[4] = {a.x, a.w, b.z, c.y};
        float ys[4] = {a.y, b.x, b.w, c.z};
        float zs[4] = {a.z, b.y, c.x, c.w};
#pragma unroll
        for (int k = 0; k < 4; ++k) {
            int bidx = b0 + ((rem + k) >= N ? 1 : 0);
            vox_one_point(xs[k], ys[k], zs[k], bidx, out);
        }
    } else {
        // Generic tail: scalar loads, no overread.
        for (int k = 0; k < 4; ++k) {
            int i = base + k;
            if (i >= npts) break;
            int bidx = b0 + ((rem + k) >= N ? 1 : 0);
            float x = pts[(size_t)3 * i + 0];
            float y = pts[(size_t)3 * i + 1];
            float z = pts[(size_t)3 * i + 2];
            vox_one_point(x, y, z, bidx, out);
        }
    }
}

extern "C" void kernel_launch(void* const* d_in, const int* in_sizes, int n_in,
                              void* d_out, int out_size, void* d_ws, size_t ws_size,
                              hipStream_t stream) {
    const float* pts = (const float*)d_in[0];
    float*       out = (float*)d_out;

    const int grid_per_batch = VX_D * VX_H * VX_W;     // 5,242,880
    int npts = in_sizes[0] / 3;                        // B*N
    int B    = out_size / grid_per_batch;              // 16
    if (B < 1) B = 1;
    int N    = npts / B;                               // 200,000

    // 1) Zero the occupancy grid with B128 stores.
    int n4 = out_size / 4;
    int zthreads = 256;                                // 8 wave32 per block
    int zblocks  = (n4 + zthreads - 1) / zthreads;
    if (zblocks < 1) zblocks = 1;
    vox_zero<<<zblocks, zthreads, 0, stream>>>(out, n4, out_size);

    // 2) Scatter valid points (4 points per thread).
    int sthreads = 256;
    int nthreads = (npts + 3) / 4;
    int sblocks  = (nthreads + sthreads - 1) / sthreads;
    if (sblocks < 1) sblocks = 1;
    vox_scatter<<<sblocks, sthreads, 0, stream>>>(pts, out, npts, N);
}